// GraphSAGE_88218628259971
// MI455X (gfx1250) — compile-verified
//
#include <hip/hip_runtime.h>
#include <math.h>

#define NN   50000
#define NE   600000
#define DIN  128
#define DHID 256
#define DOUT 128

typedef __attribute__((ext_vector_type(2))) float v2f;
typedef __attribute__((ext_vector_type(8))) float v8f;

// ---------------------------------------------------------------- utilities

__global__ void zero_kernel(float* __restrict__ p, int n) {
    int i = blockIdx.x * blockDim.x + threadIdx.x;
    int stride = gridDim.x * blockDim.x;
    for (; i < n; i += stride) p[i] = 0.0f;
}

__global__ void degree_kernel(const long long* __restrict__ dst,
                              float* __restrict__ cnt, int E) {
    int e = blockIdx.x * blockDim.x + threadIdx.x;
    if (e < E) atomicAdd(&cnt[(int)dst[e]], 1.0f);
}

// one wave (32 lanes) per edge: coalesced row gather + f32 L2 atomics scatter
__global__ void scatter_add_kernel(const float* __restrict__ x,
                                   const long long* __restrict__ src,
                                   const long long* __restrict__ dst,
                                   float* __restrict__ agg, int E, int D) {
    int gid  = blockIdx.x * blockDim.x + threadIdx.x;
    int wave = gid >> 5;
    int lane = threadIdx.x & 31;
    if (wave >= E) return;
    long long s = src[wave];
    long long d = dst[wave];
    const float* xr = x   + (size_t)s * D;
    float*       ar = agg + (size_t)d * D;
    for (int j = lane; j < D; j += 32)
        atomicAdd(&ar[j], xr[j]);
}

// --------------------------------------------------- fused dual-GEMM (WMMA)
// OUT[M,N] = act( (A1 * rowscale) @ W1 + A2 @ W2 + bias ),
// rowscale = 1/max(cnt,1)  (fuses SAGE mean into the A1 load).
//
// Block tile 128x64, 256 threads = 8 waves; wave w computes a 16x64 strip as
// four 16x16 f32 WMMA tiles (K step 4). B is staged in LDS interleaved by
// k-pairs: sB[p][2n+{0,1}] = W[2p+{0,1}][n], so a B fragment is ONE aligned
// ds_load_b64 into the WMMA source pair (no cross-register shuffles).
// Pair-row stride 160 floats == 32 banks (mod 64): lanes 0-15 (pair p) and
// lanes 16-31 (pair p+1) hit disjoint bank halves. A stride 20 floats gives
// banks {20r,20r+1} vs {20r+2,20r+3} for the two half-waves: conflict-free.
__launch_bounds__(256)
__global__ void sage_gemm_kernel(const float* __restrict__ A1,   // [M,K] neighbor sums
                                 const float* __restrict__ cnt,  // [M] degree
                                 const float* __restrict__ A2,   // [M,K] self features
                                 const float* __restrict__ W1,   // [K,N]
                                 const float* __restrict__ W2,   // [K,N]
                                 const float* __restrict__ bias, // [N]
                                 float* __restrict__ OUT,        // [M,N]
                                 int M, int K, int N, int relu) {
    __shared__ float sA1[128][20];
    __shared__ float sA2[128][20];
    __shared__ float sB1[8][160];
    __shared__ float sB2[8][160];

    const int t    = threadIdx.x;
    const int lane = t & 31;
    const int w    = t >> 5;        // wave 0..7 owns rows w*16 .. w*16+15

    const int mBlock = blockIdx.x * 128;
    const int cBlock = blockIdx.y * 64;

    // WMMA f32 16x16x4 fragment lane mapping:
    //   A: lane&15 = M row, (lane>>4)*2 = k-pair base; v.x=K+0, v.y=K+1
    //   B: lane&15 = N col, same k-pair selection
    const int frow = lane & 15;
    const int kb   = (lane >> 4) * 2;

    v8f acc[4] = {};

    // A fill mapping (two 64-row halves per chunk)
    const int ar = t >> 2;          // 0..63
    const int ac = (t & 3) * 4;     // 0,4,8,12
    const int rowA0 = mBlock + ar;
    const int rowA1 = rowA0 + 64;
    const bool ok0 = rowA0 < M;
    const bool ok1 = rowA1 < M;
    const float inv0 = ok0 ? 1.0f / fmaxf(cnt[rowA0], 1.0f) : 1.0f;
    const float inv1 = ok1 ? 1.0f / fmaxf(cnt[rowA1], 1.0f) : 1.0f;

    // B fill mapping: wave bp handles k-pair-row bp, lane covers 2 columns
    const int bp = w;               // pair-row 0..7 (k rows 2bp, 2bp+1)
    const int bn = lane * 2;        // column 0..62

    for (int kc = 0; kc < K; kc += 16) {
        // ---- stage A (rowscale fused on A1)
        float4 z = {0.f, 0.f, 0.f, 0.f};
        float4 v0 = z, v1 = z, u0 = z, u1 = z;
        if (ok0) {
            v0 = *(const float4*)(A1 + (size_t)rowA0 * K + kc + ac);
            u0 = *(const float4*)(A2 + (size_t)rowA0 * K + kc + ac);
            v0.x *= inv0; v0.y *= inv0; v0.z *= inv0; v0.w *= inv0;
        }
        if (ok1) {
            v1 = *(const float4*)(A1 + (size_t)rowA1 * K + kc + ac);
            u1 = *(const float4*)(A2 + (size_t)rowA1 * K + kc + ac);
            v1.x *= inv1; v1.y *= inv1; v1.z *= inv1; v1.w *= inv1;
        }
        *(float4*)&sA1[ar][ac]      = v0;
        *(float4*)&sA1[ar + 64][ac] = v1;
        *(float4*)&sA2[ar][ac]      = u0;
        *(float4*)&sA2[ar + 64][ac] = u1;

        // ---- stage B interleaved by k-pairs (one b128 store per matrix)
        {
            const float* r0 = W1 + (size_t)(kc + 2 * bp) * N + cBlock + bn;
            float2 x0 = *(const float2*)r0;
            float2 x1 = *(const float2*)(r0 + N);
            float4 pk = {x0.x, x1.x, x0.y, x1.y};
            *(float4*)&sB1[bp][bn * 2] = pk;

            const float* q0 = W2 + (size_t)(kc + 2 * bp) * N + cBlock + bn;
            float2 y0 = *(const float2*)q0;
            float2 y1 = *(const float2*)(q0 + N);
            float4 qk = {y0.x, y1.x, y0.y, y1.y};
            *(float4*)&sB2[bp][bn * 2] = qk;
        }

        // prefetch next chunk's A rows into cache while we crunch this one
        if (kc + 16 < K) {
            if (ok0) {
                __builtin_prefetch(A1 + (size_t)rowA0 * K + kc + 16 + ac, 0, 3);
                __builtin_prefetch(A2 + (size_t)rowA0 * K + kc + 16 + ac, 0, 3);
            }
            if (ok1) {
                __builtin_prefetch(A1 + (size_t)rowA1 * K + kc + 16 + ac, 0, 3);
                __builtin_prefetch(A2 + (size_t)rowA1 * K + kc + 16 + ac, 0, 3);
            }
        }
        __syncthreads();

        const int arow = w * 16 + frow;
#pragma unroll
        for (int ks = 0; ks < 16; ks += 4) {
            v2f a1 = *(const v2f*)&sA1[arow][ks + kb];
            v2f a2 = *(const v2f*)&sA2[arow][ks + kb];
            const int p = (ks + kb) >> 1;
#pragma unroll
            for (int nt = 0; nt < 4; ++nt) {
                const int bo = (nt * 16 + frow) * 2;
                v2f b1 = *(const v2f*)&sB1[p][bo];
                v2f b2 = *(const v2f*)&sB2[p][bo];
                acc[nt] = __builtin_amdgcn_wmma_f32_16x16x4_f32(
                    false, a1, false, b1, (short)0, acc[nt], false, false);
                acc[nt] = __builtin_amdgcn_wmma_f32_16x16x4_f32(
                    false, a2, false, b2, (short)0, acc[nt], false, false);
            }
        }
        __syncthreads();
    }

    // ---- epilogue: D layout -> VGPR r holds M=r (lanes 0-15) / r+8 (16-31)
    const int rbase = mBlock + w * 16 + ((lane >> 4) << 3);
#pragma unroll
    for (int nt = 0; nt < 4; ++nt) {
        const int col = cBlock + nt * 16 + frow;
        const float bv = bias[col];
#pragma unroll
        for (int r = 0; r < 8; ++r) {
            int row = rbase + r;
            if (row < M) {
                float v = acc[nt][r] + bv;
                if (relu) v = fmaxf(v, 0.f);
                OUT[(size_t)row * N + col] = v;
            }
        }
    }
}

// ------------------------------------------------------------- L2 normalize
__global__ void l2norm_kernel(float* __restrict__ out, int M, int D) {
    int gid  = blockIdx.x * blockDim.x + threadIdx.x;
    int row  = gid >> 5;
    int lane = threadIdx.x & 31;
    if (row >= M) return;
    float* p = out + (size_t)row * D;
    float ss = 0.f;
    for (int j = lane; j < D; j += 32) { float v = p[j]; ss += v * v; }
#pragma unroll
    for (int off = 16; off > 0; off >>= 1) ss += __shfl_xor(ss, off, 32);
    float s = 1.0f / fmaxf(sqrtf(ss), 1e-12f);
    for (int j = lane; j < D; j += 32) p[j] *= s;
}

// ------------------------------------------------------------------- driver
extern "C" void kernel_launch(void* const* d_in, const int* in_sizes, int n_in,
                              void* d_out, int out_size, void* d_ws, size_t ws_size,
                              hipStream_t stream) {
    const float*     x_orig  = (const float*)d_in[0];
    const long long* ei_orig = (const long long*)d_in[1];
    const float*     x_anon  = (const float*)d_in[2];
    const long long* ei_anon = (const long long*)d_in[3];
    const float*     W_l1    = (const float*)d_in[4];
    const float*     W_r1    = (const float*)d_in[5];
    const float*     b1      = (const float*)d_in[6];
    const float*     W_l2    = (const float*)d_in[7];
    const float*     W_r2    = (const float*)d_in[8];
    const float*     b2      = (const float*)d_in[9];

    // workspace layout (floats): cnt | aggbuf (agg1 then agg2) | h
    float* cnt    = (float*)d_ws;
    float* aggbuf = cnt + 50176;                 // 256-aligned past NN
    float* h      = aggbuf + (size_t)NN * DHID;

    for (int g = 0; g < 2; ++g) {
        const float*     x   = g ? x_anon : x_orig;
        const long long* src = g ? ei_anon : ei_orig;
        const long long* dst = src + NE;
        float* outg = (float*)d_out + (size_t)g * NN * DOUT;

        // ---- layer 1: mean-aggregate x, then relu(agg@W_l1 + x@W_r1 + b1)
        zero_kernel<<<2048, 256, 0, stream>>>(cnt, 50176 + NN * DIN);
        degree_kernel<<<(NE + 255) / 256, 256, 0, stream>>>(dst, cnt, NE);
        scatter_add_kernel<<<(NE * 32 + 255) / 256, 256, 0, stream>>>(
            x, src, dst, aggbuf, NE, DIN);
        {
            dim3 grid((NN + 127) / 128, DHID / 64);
            sage_gemm_kernel<<<grid, 256, 0, stream>>>(
                aggbuf, cnt, x, W_l1, W_r1, b1, h, NN, DIN, DHID, 1);
        }

        // ---- layer 2: mean-aggregate h, then agg2@W_l2 + h@W_r2 + b2
        zero_kernel<<<4096, 256, 0, stream>>>(aggbuf, NN * DHID);
        scatter_add_kernel<<<(NE * 32 + 255) / 256, 256, 0, stream>>>(
            h, src, dst, aggbuf, NE, DHID);
        {
            dim3 grid((NN + 127) / 128, DOUT / 64);
            sage_gemm_kernel<<<grid, 256, 0, stream>>>(
                aggbuf, cnt, h, W_l2, W_r2, b2, outg, NN, DHID, DOUT, 0);
        }

        // ---- row-wise L2 normalize
        l2norm_kernel<<<(NN * 32 + 255) / 256, 256, 0, stream>>>(outg, NN, DOUT);
    }
}